// GATEncoder_17059610100458
// MI455X (gfx1250) — compile-verified
//
#include <hip/hip_runtime.h>

#define N_NODES 100000
#define N_EDGES 1600000
#define IN_CH   128
#define HEADS   4
#define CPH     32
#define HC      128            // HEADS*CPH
#define SLOPE   0.2f

typedef __attribute__((ext_vector_type(2))) float v2f;
typedef __attribute__((ext_vector_type(8))) float v8f;

// ---- order-preserving float<->uint encoding for atomic max on f32 ----
__device__ __forceinline__ unsigned enc_f(float x) {
    int i = __float_as_int(x);
    return (i >= 0) ? ((unsigned)i | 0x80000000u) : ~(unsigned)i;
}
__device__ __forceinline__ float dec_f(unsigned u) {
    int i = (u & 0x80000000u) ? (int)(u & 0x7FFFFFFFu) : (int)~u;
    return __int_as_float(i);
}

// ============================================================
// 1) h = x @ W via V_WMMA_F32_16X16X4_F32 (one 16x16 tile / wave32)
// ============================================================
__global__ __launch_bounds__(256) void gat_proj_wmma(const float* __restrict__ x,
                                                     const float* __restrict__ W,
                                                     float* __restrict__ h) {
    const int lane  = threadIdx.x & 31;
    const int wave  = threadIdx.x >> 5;
    const int tile  = blockIdx.x * 8 + wave;     // 6250 blocks * 8 waves = 50000 tiles
    const int tm    = tile >> 3;                 // 0..6249  (M tile, 16 rows)
    const int tn    = tile & 7;                  // 0..7     (N tile, 16 cols)
    const int m16   = lane & 15;
    const int khalf = (lane >> 4) << 1;          // 0 (lanes 0-15) or 2 (lanes 16-31)
    const int mrow  = (tm << 4) + m16;
    const int ncol  = (tn << 4) + m16;

    const float* __restrict__ xrow = x + (size_t)mrow * IN_CH;

    v8f acc = {};
#pragma unroll
    for (int k0 = 0; k0 < IN_CH; k0 += 4) {
        v2f a, b;
        // A 16x4 f32 layout: VGPR0=K (khalf), VGPR1=K+1, lanes 0-15 K0/K1, lanes 16-31 K2/K3
        a.x = xrow[k0 + khalf + 0];
        a.y = xrow[k0 + khalf + 1];
        // B 4x16 f32 layout mirrored: row K striped across lanes within a VGPR
        b.x = W[(size_t)(k0 + khalf + 0) * HC + ncol];
        b.y = W[(size_t)(k0 + khalf + 1) * HC + ncol];
        acc = __builtin_amdgcn_wmma_f32_16x16x4_f32(false, a, false, b,
                                                    (short)0, acc, false, false);
    }

    // C/D layout: VGPR v -> M = v (lanes 0-15) or v+8 (lanes 16-31), N = lane&15
    const int rbase = (tm << 4) + ((lane >> 4) << 3);
#pragma unroll
    for (int v = 0; v < 8; ++v) {
        h[(size_t)(rbase + v) * HC + ncol] = acc[v];
    }
}

// ============================================================
// 2) per-(node,head) attention logits + init of max/denom buffers
// ============================================================
__global__ __launch_bounds__(256) void gat_attn(const float* __restrict__ h,
                                                const float* __restrict__ att_src,
                                                const float* __restrict__ att_dst,
                                                float* __restrict__ a_src,
                                                float* __restrict__ a_dst,
                                                unsigned* __restrict__ m_enc,
                                                float* __restrict__ denom,
                                                int nh) {
    int i = blockIdx.x * blockDim.x + threadIdx.x;
    if (i >= nh) return;
    const int hd = i & (HEADS - 1);
    const float* hp = h + (size_t)i * CPH;   // i = n*HEADS+hd, h[n][hd][c] contiguous
    const float* as = att_src + hd * CPH;
    const float* ad = att_dst + hd * CPH;
    float s = 0.f, d = 0.f;
#pragma unroll
    for (int c = 0; c < CPH; ++c) {
        float v = hp[c];
        s = fmaf(v, as[c], s);
        d = fmaf(v, ad[c], d);
    }
    a_src[i] = s;
    a_dst[i] = d;
    m_enc[i] = 0u;        // below enc() of any float -> acts as -inf
    denom[i] = 0.f;
}

// ============================================================
// 3) out[i] = bias[i % 128]  (seed for scatter-add)
// ============================================================
__global__ __launch_bounds__(256) void gat_init_out(float* __restrict__ out,
                                                    const float* __restrict__ bias,
                                                    int total) {
    int i = blockIdx.x * blockDim.x + threadIdx.x;
    if (i < total) out[i] = bias[i & (HC - 1)];
}

// ============================================================
// 4) segment max over destination: atomicMax on order-preserving u32
// ============================================================
__global__ __launch_bounds__(256) void gat_edge_max(const int* __restrict__ src,
                                                    const int* __restrict__ dst,
                                                    const float* __restrict__ a_src,
                                                    const float* __restrict__ a_dst,
                                                    unsigned* __restrict__ m_enc) {
    int i = blockIdx.x * blockDim.x + threadIdx.x;   // over E*HEADS (exact)
    int e  = i >> 2;
    int hd = i & 3;
    int s = src[e], d = dst[e];
    float v = a_src[s * HEADS + hd] + a_dst[d * HEADS + hd];
    v = (v > 0.f) ? v : v * SLOPE;
    atomicMax(&m_enc[d * HEADS + hd], enc_f(v));
}

// ============================================================
// 5) denom[dst,h] += exp(e - max)
// ============================================================
__global__ __launch_bounds__(256) void gat_edge_denom(const int* __restrict__ src,
                                                      const int* __restrict__ dst,
                                                      const float* __restrict__ a_src,
                                                      const float* __restrict__ a_dst,
                                                      const unsigned* __restrict__ m_enc,
                                                      float* __restrict__ denom) {
    int i = blockIdx.x * blockDim.x + threadIdx.x;   // over E*HEADS (exact)
    int e  = i >> 2;
    int hd = i & 3;
    int s = src[e], d = dst[e];
    int nhd = d * HEADS + hd;
    float v = a_src[s * HEADS + hd] + a_dst[nhd];
    v = (v > 0.f) ? v : v * SLOPE;
    float p = __expf(v - dec_f(m_enc[nhd]));
    atomicAdd(&denom[nhd], p);
}

// ============================================================
// 6) out[dst] += alpha * h[src] : one wave32 per edge, float4/lane
// ============================================================
__global__ __launch_bounds__(256) void gat_aggregate(const int* __restrict__ src,
                                                     const int* __restrict__ dst,
                                                     const float* __restrict__ h,
                                                     const float* __restrict__ a_src,
                                                     const float* __restrict__ a_dst,
                                                     const unsigned* __restrict__ m_enc,
                                                     const float* __restrict__ denom,
                                                     float* __restrict__ out) {
    const int lane = threadIdx.x & 31;
    const int e    = (blockIdx.x << 3) + (threadIdx.x >> 5);  // 200000 blocks * 8 waves (exact)
    const int s = src[e], d = dst[e];
    const int hd  = lane >> 3;                // 4 heads x 8 lanes
    const int nhd = d * HEADS + hd;

    float a = a_src[s * HEADS + hd] + a_dst[nhd];
    a = (a > 0.f) ? a : a * SLOPE;
    float p     = __expf(a - dec_f(m_enc[nhd]));
    float alpha = p / (denom[nhd] + 1e-16f);

    // lane*4 == hd*32 + (lane&7)*4 : exact channel base
    const float4 hv = *(const float4*)(h + (size_t)s * HC + (lane << 2));
    float* op = out + (size_t)d * HC + (lane << 2);
    atomicAdd(op + 0, hv.x * alpha);
    atomicAdd(op + 1, hv.y * alpha);
    atomicAdd(op + 2, hv.z * alpha);
    atomicAdd(op + 3, hv.w * alpha);
}

// ============================================================
extern "C" void kernel_launch(void* const* d_in, const int* in_sizes, int n_in,
                              void* d_out, int out_size, void* d_ws, size_t ws_size,
                              hipStream_t stream) {
    const float* x       = (const float*)d_in[0];          // [N, IN]
    const int*   ei      = (const int*)  d_in[1];          // [2, E] flat
    const float* W       = (const float*)d_in[2];          // [IN, H*C]
    const float* att_src = (const float*)d_in[3];          // [H, C]
    const float* att_dst = (const float*)d_in[4];          // [H, C]
    const float* bias    = (const float*)d_in[5];          // [H*C]
    float*       out     = (float*)d_out;                  // [N, H*C]

    const int* src = ei;             // row 0
    const int* dst = ei + N_EDGES;   // row 1

    // workspace carve-out (256B aligned)
    char* ws = (char*)d_ws;
    size_t off = 0;
    auto carve = [&](size_t bytes) { void* p = ws + off; off += (bytes + 255) & ~(size_t)255; return p; };
    float*    h      = (float*)   carve((size_t)N_NODES * HC * sizeof(float));     // 51.2 MB
    float*    a_src  = (float*)   carve((size_t)N_NODES * HEADS * sizeof(float));  // 1.6 MB
    float*    a_dst  = (float*)   carve((size_t)N_NODES * HEADS * sizeof(float));
    unsigned* m_enc  = (unsigned*)carve((size_t)N_NODES * HEADS * sizeof(unsigned));
    float*    denom  = (float*)   carve((size_t)N_NODES * HEADS * sizeof(float));

    // 1) projection (WMMA): 6250 blocks x 8 waves, 16x16 tile per wave
    gat_proj_wmma<<<6250, 256, 0, stream>>>(x, W, h);

    // 2) attention logits + buffer init
    const int nh = N_NODES * HEADS;
    gat_attn<<<(nh + 255) / 256, 256, 0, stream>>>(h, att_src, att_dst,
                                                   a_src, a_dst, m_enc, denom, nh);

    // 3) seed output with bias
    const int total = N_NODES * HC;                         // 12.8M, divides 256
    gat_init_out<<<total / 256, 256, 0, stream>>>(out, bias, total);

    // 4) segment max   (E*H = 6.4M, divides 256)
    gat_edge_max<<<(N_EDGES * HEADS) / 256, 256, 0, stream>>>(src, dst, a_src, a_dst, m_enc);

    // 5) softmax denominator
    gat_edge_denom<<<(N_EDGES * HEADS) / 256, 256, 0, stream>>>(src, dst, a_src, a_dst,
                                                                m_enc, denom);

    // 6) weighted scatter-add aggregation: one wave per edge
    gat_aggregate<<<N_EDGES / 8, 256, 0, stream>>>(src, dst, h, a_src, a_dst,
                                                   m_enc, denom, out);
}